// TopKGateDynamic_5025111736592
// MI455X (gfx1250) — compile-verified
//
#include <hip/hip_runtime.h>
#include <math.h>

#define S_    8192
#define D_    1024
#define E_    64
#define CAP_  256
#define NEG_INF (-3.402823466e+38f)
#define EPS_F 1.1920929e-07f

typedef __attribute__((ext_vector_type(2))) float v2f;
typedef __attribute__((ext_vector_type(8))) float v8f;

// ---------------------------------------------------------------------------
// monotonic uint key for float ordering (matches top_k ordering incl. negatives)
__device__ __forceinline__ unsigned fkey(float v) {
  unsigned u = __float_as_uint(v);
  return (u & 0x80000000u) ? ~u : (u | 0x80000000u);
}

// ---------------------------------------------------------------------------
// Kernel 0: zero per-expert accumulators (must run every launch: deterministic)
__global__ void init_kernel(float* me_sum, int* cnt) {
  int e = threadIdx.x;
  if (e < E_) { me_sum[e] = 0.0f; cnt[e] = 0; }
}

// ---------------------------------------------------------------------------
// Kernel 1: logits = x @ W^T via V_WMMA_F32_16X16X4_F32.
// One wave per 16-row panel, covering all 64 expert columns (4 N-tiles).
__device__ __forceinline__ v8f wmma4(v2f a, v2f b, v8f c) {
  return __builtin_amdgcn_wmma_f32_16x16x4_f32(
      /*neg_a=*/false, a, /*neg_b=*/false, b,
      /*c_mod=*/(short)0, c, /*reuse_a=*/false, /*reuse_b=*/false);
}

__global__ void __launch_bounds__(256) gemm_kernel(const float* __restrict__ x,
                                                   const float* __restrict__ W,
                                                   float* __restrict__ logits) {
  const int lane = threadIdx.x & 31;
  const int wave = threadIdx.x >> 5;
  const int m0   = (blockIdx.x * 8 + wave) * 16;
  const int l16  = lane & 15;   // M (for A) / N (for B) within tile
  const int kg   = lane >> 4;   // K-pair group: 0 -> K{0,1}, 1 -> K{2,3}

  // A: 16x4 f32 fragment.  lane(l16,kg) holds x[m0+l16, k0+2kg .. +1]
  const float* arow = x + (size_t)(m0 + l16) * D_ + 2 * kg;
  // B: 4x16 f32 fragment of W^T.  lane(l16,kg) holds W[col, k0+2kg .. +1]
  const float* bcol = W + (size_t)l16 * D_ + 2 * kg;

  v8f acc0 = {}, acc1 = {}, acc2 = {}, acc3 = {};
#pragma unroll 4
  for (int k0 = 0; k0 < D_; k0 += 4) {
    v2f a  = *(const v2f*)(arow + k0);
    v2f b0 = *(const v2f*)(bcol + k0);
    v2f b1 = *(const v2f*)(bcol + (size_t)16 * D_ + k0);
    v2f b2 = *(const v2f*)(bcol + (size_t)32 * D_ + k0);
    v2f b3 = *(const v2f*)(bcol + (size_t)48 * D_ + k0);
    acc0 = wmma4(a, b0, acc0);
    acc1 = wmma4(a, b1, acc1);
    acc2 = wmma4(a, b2, acc2);
    acc3 = wmma4(a, b3, acc3);
  }

  // C/D layout: VGPR r -> M = r (lanes 0-15) / M = r+8 (lanes 16-31), N = lane%16
  const int nb = l16;
  const int mh = kg;
#pragma unroll
  for (int r = 0; r < 8; ++r) {
    int row = m0 + mh * 8 + r;
    float* o = logits + (size_t)row * E_ + nb;
    o[0]  = acc0[r];
    o[16] = acc1[r];
    o[32] = acc2[r];
    o[48] = acc3[r];
  }
}

// ---------------------------------------------------------------------------
// Kernel 2: per-token softmax stats, top-2 (jax tie rule: lower index wins),
// masked-logit keys (transposed, coalesced per expert), me/exp_counts accum.
__global__ void __launch_bounds__(256) row_kernel(const float* __restrict__ logits,
                                                  float* __restrict__ rowmax,
                                                  float* __restrict__ rowden,
                                                  int* __restrict__ top2,
                                                  int* __restrict__ loc2,
                                                  unsigned* __restrict__ keys,
                                                  float* __restrict__ me_sum,
                                                  int* __restrict__ cnt) {
  __shared__ float sMe[E_];
  __shared__ int   sCnt[E_];
  const int tid = threadIdx.x;
  if (tid < E_) { sMe[tid] = 0.0f; sCnt[tid] = 0; }
  __syncthreads();

  const int s = blockIdx.x * 256 + tid;
  const float* row = logits + (size_t)s * E_;

  float v1 = NEG_INF, v2 = NEG_INF, mx = NEG_INF;
  int i1 = -1, i2 = -1;
  for (int e = 0; e < E_; ++e) {
    float v = row[e];
    mx = fmaxf(mx, v);
    if (v > v1)      { v2 = v1; i2 = i1; v1 = v; i1 = e; }
    else if (v > v2) { v2 = v;  i2 = e; }
  }
  float den = 0.0f;
  for (int e = 0; e < E_; ++e) den += expf(row[e] - mx);

  rowmax[s] = mx;
  rowden[s] = den;
  top2[2 * s]     = i1;
  top2[2 * s + 1] = i2;
  loc2[2 * s]     = -1;
  loc2[2 * s + 1] = -1;

  for (int e = 0; e < E_; ++e) {
    float g = expf(row[e] - mx) / den;
    atomicAdd(&sMe[e], g);
    float masked = (e == i1 || e == i2) ? row[e] : 0.0f;
    keys[(size_t)e * S_ + s] = fkey(masked);   // contiguous in s: coalesced
  }
  atomicAdd(&sCnt[i1], 1);
  atomicAdd(&sCnt[i2], 1);
  __syncthreads();
  if (tid < E_) {
    atomicAdd(&me_sum[tid], sMe[tid]);
    atomicAdd(&cnt[tid], sCnt[tid]);
  }
}

// ---------------------------------------------------------------------------
// Kernel 3: one block per expert. Bitwise radix-select of the CAP_-th largest
// key over all S_ column values, then chunked scan assigning capacity slots
// with exact tie-by-lower-index semantics; writes locations = cumsum(mask)-1.
__global__ void __launch_bounds__(256) select_kernel(const unsigned* __restrict__ keys,
                                                     const int* __restrict__ top2,
                                                     int* __restrict__ loc2) {
  const int e   = blockIdx.x;
  const int tid = threadIdx.x;
  const unsigned* col = keys + (size_t)e * S_;

  __shared__ int red[256];
  __shared__ int sB[256];
  __shared__ unsigned sPref;
  __shared__ int sNeed;
  if (tid == 0) { sPref = 0u; sNeed = CAP_; }
  __syncthreads();

  // --- radix select: find CAP_-th largest key T ---
  for (int b = 31; b >= 0; --b) {
    unsigned trial = sPref | (1u << b);
    int c = 0;
    for (int s = tid; s < S_; s += 256) {
      if ((col[s] >> b) == (trial >> b)) ++c;
    }
    red[tid] = c;
    __syncthreads();
    for (int off = 128; off > 0; off >>= 1) {
      if (tid < off) red[tid] += red[tid + off];
      __syncthreads();
    }
    if (tid == 0) {
      if (red[0] >= sNeed) sPref = trial;
      else                 sNeed -= red[0];
    }
    __syncthreads();
  }
  const unsigned T = sPref;

  // --- count strictly greater than T ---
  {
    int c = 0;
    for (int s = tid; s < S_; s += 256) if (col[s] > T) ++c;
    red[tid] = c;
    __syncthreads();
    for (int off = 128; off > 0; off >>= 1) {
      if (tid < off) red[tid] += red[tid + off];
      __syncthreads();
    }
  }
  const int remaining = CAP_ - red[0];   // tie slots (ties broken by index)
  __syncthreads();

  // --- sequential chunked scan over tokens: slot assignment + cumsum ---
  int runT = 0, runK = 0;
  for (int chunk = 0; chunk < S_ / 256; ++chunk) {
    const int s = chunk * 256 + tid;
    const unsigned k = col[s];
    const int t0 = top2[2 * s], t1 = top2[2 * s + 1];
    const int selected = (t0 == e) || (t1 == e);
    const int isTie = (k == T) ? 1 : 0;

    // inclusive scan of ties (ALL tokens with key==T consume tie slots)
    red[tid] = isTie;
    __syncthreads();
    for (int off = 1; off < 256; off <<= 1) {
      int t = (tid >= off) ? red[tid - off] : 0;
      __syncthreads();
      red[tid] += t;
      __syncthreads();
    }
    const int tieRank = runT + red[tid] - isTie;
    const int tieTot  = red[255];

    const int kept =
        (selected && (k > T || (isTie && tieRank < remaining))) ? 1 : 0;

    red[tid] = 0;  // unused; keep red stable
    sB[tid] = kept;
    __syncthreads();
    for (int off = 1; off < 256; off <<= 1) {
      int t = (tid >= off) ? sB[tid - off] : 0;
      __syncthreads();
      sB[tid] += t;
      __syncthreads();
    }
    if (kept) {
      int j = (t0 == e) ? 0 : 1;
      loc2[2 * s + j] = runK + sB[tid] - 1;   // cumsum(mask)-1
    }
    runT += tieTot;
    runK += sB[255];
    __syncthreads();
  }
}

// ---------------------------------------------------------------------------
// Kernel 4: l_aux = mean(me*ce) * E*E/K ; exp_counts tail.
__global__ void laux_kernel(const float* __restrict__ me_sum,
                            const int* __restrict__ cnt,
                            float* __restrict__ out) {
  __shared__ float red[E_];
  const int e = threadIdx.x;
  red[e] = me_sum[e] * (float)cnt[e];
  __syncthreads();
  for (int off = 32; off > 0; off >>= 1) {
    if (e < off) red[e] += red[e + off];
    __syncthreads();
  }
  if (e == 0) {
    // sum(me_sum*cnt) * (E/K) / S^2
    out[0] = red[0] * ((float)E_ / 2.0f) / ((float)S_ * (float)S_);
  }
  out[1 + 2 * (size_t)S_ * E_ * CAP_ + e] = (float)cnt[e];
}

// ---------------------------------------------------------------------------
// Kernel 5: scatter combine_weights and dispatch_mask (sparse nonzeros only).
__global__ void __launch_bounds__(256) combine_kernel(const float* __restrict__ logits,
                                                      const float* __restrict__ rowmax,
                                                      const float* __restrict__ rowden,
                                                      const int* __restrict__ top2,
                                                      const int* __restrict__ loc2,
                                                      float* __restrict__ out) {
  const int s = blockIdx.x * 256 + threadIdx.x;
  const float mx = rowmax[s];
  const float den = rowden[s];

  int   e[2], lc[2];
  float g[2];
  float denom = 0.0f;
#pragma unroll
  for (int j = 0; j < 2; ++j) {
    e[j]  = top2[2 * s + j];
    lc[j] = loc2[2 * s + j];
    g[j]  = expf(logits[(size_t)s * E_ + e[j]] - mx) / den;
    if (lc[j] >= 0) denom += g[j];
  }
  denom = fmaxf(denom, EPS_F);

  float* cw = out + 1;
  float* dm = out + 1 + (size_t)S_ * E_ * CAP_;
#pragma unroll
  for (int j = 0; j < 2; ++j) {
    if (lc[j] >= 0) {
      size_t idx = ((size_t)s * E_ + e[j]) * CAP_ + lc[j];
      cw[idx] = g[j] / denom;
      dm[idx] = 1.0f;
    }
  }
}

// ---------------------------------------------------------------------------
extern "C" void kernel_launch(void* const* d_in, const int* in_sizes, int n_in,
                              void* d_out, int out_size, void* d_ws, size_t ws_size,
                              hipStream_t stream) {
  const float* x = (const float*)d_in[0];   // (S, D) fp32
  const float* W = (const float*)d_in[1];   // (E, D) fp32
  float* out = (float*)d_out;

  // workspace carve (~4.3 MB)
  float*    logits = (float*)d_ws;                         // S*E
  unsigned* keys   = (unsigned*)(logits + (size_t)S_ * E_); // S*E
  float*    rowmax = (float*)(keys + (size_t)S_ * E_);      // S
  float*    rowden = rowmax + S_;                           // S
  int*      top2   = (int*)(rowden + S_);                   // 2S
  int*      loc2   = top2 + 2 * S_;                         // 2S
  float*    me_sum = (float*)(loc2 + 2 * S_);               // E
  int*      cnt    = (int*)(me_sum + E_);                   // E

  // zero the (mostly sparse) 1 GB output region in one bulk memset
  hipMemsetAsync(d_out, 0, (size_t)out_size * sizeof(float), stream);

  init_kernel<<<1, 64, 0, stream>>>(me_sum, cnt);
  gemm_kernel<<<S_ / 128, 256, 0, stream>>>(x, W, logits);
  row_kernel<<<S_ / 256, 256, 0, stream>>>(logits, rowmax, rowden, top2, loc2,
                                           keys, me_sum, cnt);
  select_kernel<<<E_, 256, 0, stream>>>(keys, top2, loc2);
  laux_kernel<<<1, E_, 0, stream>>>(me_sum, cnt, out);
  combine_kernel<<<S_ / 256, 256, 0, stream>>>(logits, rowmax, rowden, top2,
                                               loc2, out);
}